// EdgeConv_74741020885664
// MI455X (gfx1250) — compile-verified
//
#include <hip/hip_runtime.h>

typedef __attribute__((ext_vector_type(2))) float v2f;
typedef __attribute__((ext_vector_type(8))) float v8f;

#define NTOT 4096
#define BATCH 8
#define CIN 64
#define KNN 20
#define OCH 64

static __device__ __forceinline__ v8f wmma_f32_16x16x4(v2f a, v2f b, v8f c) {
  // D = A(16x4) * B(4x16) + C, fp32 matrix core
  return __builtin_amdgcn_wmma_f32_16x16x4_f32(false, a, false, b, (short)0, c, false, false);
}

// ---------------------------------------------------------------------------
// Kernel 0: x (B,C,N) -> xt (B,N,C) + candidate squared norms xx (B,N)
// ---------------------------------------------------------------------------
__global__ __launch_bounds__(256) void k0_transpose_norm(const float* __restrict__ x,
                                                         float* __restrict__ xt,
                                                         float* __restrict__ xx) {
  const int n = blockIdx.x * 256 + threadIdx.x;
  const int b = blockIdx.y;
  const float* src = x + (size_t)b * CIN * NTOT + n;
  float* dst = xt + ((size_t)(b * NTOT + n)) * CIN;
  float nrm = 0.f;
#pragma unroll
  for (int c = 0; c < CIN; c += 4) {
    float v0 = src[(size_t)(c + 0) * NTOT];
    float v1 = src[(size_t)(c + 1) * NTOT];
    float v2 = src[(size_t)(c + 2) * NTOT];
    float v3 = src[(size_t)(c + 3) * NTOT];
    nrm += v0 * v0 + v1 * v1 + v2 * v2 + v3 * v3;
    float4 st = make_float4(v0, v1, v2, v3);
    *(float4*)(dst + c) = st;
  }
  xx[b * NTOT + n] = nrm;
}

// ---------------------------------------------------------------------------
// Kernel 1: fused WMMA pairwise-inner-product + streaming top-20 selection.
// Selection key: 2*inner - ||cand||^2  (query norm is a per-row constant).
// WG = 256 threads = 8 waves; 64 queries/WG; candidate chunks of 64,
// double-buffered through registers; query A-fragments live in VGPRs.
// ---------------------------------------------------------------------------
__global__ __launch_bounds__(256) void k1_knn(const float* __restrict__ xt,
                                              const float* __restrict__ xx,
                                              int* __restrict__ idx) {
  __shared__ float smem[10368];
  float (*Cs)[68] = (float(*)[68])smem;               // 64x68
  float (*Ds)[68] = (float(*)[68])(smem + 64 * 68);   // 64x68
  float* xcn = smem + 2 * 64 * 68;                    // [2][64] double-buffered

  const int tid = threadIdx.x;
  const int b = blockIdx.y;
  const int q0 = blockIdx.x * 64;
  const int wave = tid >> 5;
  const int lane = tid & 31;
  const int ml = lane & 15;
  const int kh = (lane >> 4) << 1;  // lanes 16-31 hold K+2/K+3 of each fragment
  const size_t bN = (size_t)b * NTOT;

  const int crow = tid >> 2;        // staging row (0..63)
  const int ccol = (tid & 3) * 16;  // staging col base

  const int mt = wave >> 1;         // query tile (2 waves share it)
  const int nt0 = (wave & 1) * 2;   // this wave's 2 candidate tiles

  // query A-fragments: registers, loaded once from global
  v2f af[16];
  {
    const float* qrow = xt + (bN + q0 + mt * 16 + ml) * CIN + kh;
#pragma unroll
    for (int s = 0; s < 16; ++s) af[s] = *(const v2f*)(qrow + 4 * s);
  }

  // per-thread register top-20 (sorted desc, fully static indexing)
  const int srow = tid >> 2;
  const int ssub = tid & 3;
  float dl[KNN];
  int il[KNN];
#pragma unroll
  for (int j = 0; j < KNN; ++j) { dl[j] = -3.4e38f; il[j] = 0; }

  // prologue: stage + store chunk 0
  float4 stg[4];
  {
    const float* src = xt + (bN + crow) * CIN + ccol;
#pragma unroll
    for (int u = 0; u < 4; ++u) stg[u] = *(const float4*)(src + 4 * u);
  }
  float xstg = (tid < 64) ? xx[bN + tid] : 0.f;
#pragma unroll
  for (int u = 0; u < 4; ++u) *(float4*)(&Cs[crow][ccol + 4 * u]) = stg[u];
  if (tid < 64) xcn[tid] = xstg;
  __syncthreads();

#pragma unroll 1
  for (int m0 = 0; m0 < NTOT; m0 += 64) {
    const int buf = (m0 >> 6) & 1;
    const bool more = (m0 + 64) < NTOT;

    // issue next chunk's global loads early (latency hidden by WMMA below)
    if (more) {
      const float* src = xt + (bN + m0 + 64 + crow) * CIN + ccol;
#pragma unroll
      for (int u = 0; u < 4; ++u) stg[u] = *(const float4*)(src + 4 * u);
      if (tid < 64) xstg = xx[bN + m0 + 64 + tid];
    }

    v8f acc0 = {0, 0, 0, 0, 0, 0, 0, 0};
    v8f acc1 = {0, 0, 0, 0, 0, 0, 0, 0};
#pragma unroll
    for (int s = 0; s < 16; ++s) {
      int k0 = 4 * s + kh;
      v2f b0 = {Cs[nt0 * 16 + ml][k0], Cs[nt0 * 16 + ml][k0 + 1]};
      v2f b1 = {Cs[(nt0 + 1) * 16 + ml][k0], Cs[(nt0 + 1) * 16 + ml][k0 + 1]};
      acc0 = wmma_f32_16x16x4(af[s], b0, acc0);
      acc1 = wmma_f32_16x16x4(af[s], b1, acc1);
    }
    // D tiles -> LDS (lane<16: rows 0-7, lane>=16: rows 8-15 of the tile)
    int rbase = mt * 16 + ((lane >> 4) << 3);
#pragma unroll
    for (int i = 0; i < 8; ++i) {
      Ds[rbase + i][nt0 * 16 + ml] = acc0[i];
      Ds[rbase + i][(nt0 + 1) * 16 + ml] = acc1[i];
    }
    __syncthreads();  // Cs reads done, Ds ready

    // store staged next chunk into Cs (not read during selection)
    if (more) {
#pragma unroll
      for (int u = 0; u < 4; ++u) *(float4*)(&Cs[crow][ccol + 4 * u]) = stg[u];
      if (tid < 64) xcn[(buf ^ 1) * 64 + tid] = xstg;
    }

    // streaming top-20: 4 threads/row, 16 candidates each
    const float* xc = xcn + buf * 64;
#pragma unroll 1
    for (int j = 0; j < 16; ++j) {
      int col = ssub * 16 + j;
      float d = 2.f * Ds[srow][col] - xc[col];
      if (d > dl[KNN - 1]) {
        dl[KNN - 1] = d;
        il[KNN - 1] = m0 + col;
#pragma unroll
        for (int t = KNN - 1; t > 0; --t) {
          if (dl[t] > dl[t - 1]) {
            float td = dl[t]; dl[t] = dl[t - 1]; dl[t - 1] = td;
            int ti = il[t]; il[t] = il[t - 1]; il[t - 1] = ti;
          }
        }
      }
    }
    __syncthreads();  // selection done before next Ds overwrite
  }

  // merge 4 partial lists per row (reuse smem; all buffers dead now)
  float* mrg = smem;
#pragma unroll
  for (int j = 0; j < KNN; ++j) {
    mrg[srow * 160 + ssub * KNN + j] = dl[j];
    mrg[srow * 160 + 80 + ssub * KNN + j] = __int_as_float(il[j]);
  }
  __syncthreads();
  if (tid < 64) {
    float* md = mrg + tid * 160;
    int* op = idx + ((size_t)(bN + q0 + tid)) * KNN;
    for (int j = 0; j < KNN; ++j) {
      float best = -3.4e38f;
      int bs = 0;
      for (int s2 = 0; s2 < 80; ++s2)
        if (md[s2] > best) { best = md[s2]; bs = s2; }
      op[j] = __float_as_int(md[80 + bs]);
      md[bs] = -3.4e38f;
    }
  }
}

// ---------------------------------------------------------------------------
// Kernel 2: gather + edge GEMM (WMMA f32) + BN + LeakyReLU + max over K.
// WG = 256 threads = 8 waves; 8 points/WG -> 160 rows = 10 M-tiles exactly.
// Wave w: O-tile (w&3), row half (w>>2)*80 (= exactly 4 points).
// ---------------------------------------------------------------------------
__global__ __launch_bounds__(256) void k2_edgeconv(const float* __restrict__ xt,
                                                   const int* __restrict__ idx,
                                                   const float* __restrict__ Wm,
                                                   const float* __restrict__ gamma,
                                                   const float* __restrict__ beta,
                                                   const float* __restrict__ rmean,
                                                   const float* __restrict__ rvar,
                                                   float* __restrict__ out) {
  __shared__ float E1[160][68];  // feat - center  (first 64 K-columns)
  __shared__ float Ce[8][64];    // centers        (second 64 K-columns)
  __shared__ int nbr[160];
  __shared__ float ymax[64][9];

  const int tid = threadIdx.x;
  const int b = blockIdx.y;
  const int n0 = blockIdx.x * 8;
  const int wave = tid >> 5;
  const int lane = tid & 31;
  const int ml = lane & 15;
  const int kh = (lane >> 4) << 1;
  const int h = lane >> 4;
  const size_t bN = (size_t)b * NTOT;

  if (tid < 160) {
    int p = tid / KNN, kk = tid - p * KNN;
    nbr[tid] = idx[(bN + n0 + p) * KNN + kk];
  }
#pragma unroll
  for (int e = 0; e < 2; ++e) {
    int flat = e * 256 + tid;
    int p = flat >> 6, c = flat & 63;
    Ce[p][c] = xt[(bN + n0 + p) * CIN + c];
  }

  // W^T fragments held in registers: B[k][n] = W[o0+ml][k]
  const int o0 = (wave & 3) * 16;
  const int mhalf = wave >> 2;
  v2f wf[32];
  const float* wrow = Wm + (o0 + ml) * 128 + kh;
#pragma unroll
  for (int s = 0; s < 32; ++s) wf[s] = *(const v2f*)(wrow + 4 * s);

  __syncthreads();  // nbr, Ce ready

  // gather: stage 40 floats/thread through registers, then subtract + store
  float4 stg[10];
#pragma unroll
  for (int e = 0; e < 5; ++e) {
    int seg = e * 256 + tid;
    int r = seg >> 3, c = (seg & 7) * 8;
    const float* src = xt + (bN + nbr[r]) * CIN + c;
    stg[2 * e] = *(const float4*)(src);
    stg[2 * e + 1] = *(const float4*)(src + 4);
  }
#pragma unroll
  for (int e = 0; e < 5; ++e) {
    int seg = e * 256 + tid;
    int r = seg >> 3, c = (seg & 7) * 8;
    int p = r / KNN;
    float4 c0 = *(const float4*)(&Ce[p][c]);
    float4 c1 = *(const float4*)(&Ce[p][c + 4]);
    float4 a0 = stg[2 * e], a1 = stg[2 * e + 1];
    a0.x -= c0.x; a0.y -= c0.y; a0.z -= c0.z; a0.w -= c0.w;
    a1.x -= c1.x; a1.y -= c1.y; a1.z -= c1.z; a1.w -= c1.w;
    *(float4*)(&E1[r][c]) = a0;
    *(float4*)(&E1[r][c + 4]) = a1;
  }
  __syncthreads();

  v8f acc[5];
#pragma unroll
  for (int t = 0; t < 5; ++t) acc[t] = (v8f){0, 0, 0, 0, 0, 0, 0, 0};

  const int rb = mhalf * 80;
  int pl[5];
#pragma unroll
  for (int t = 0; t < 5; ++t) pl[t] = (rb + t * 16 + ml) / KNN;

// K in [0,64): edge = feat - center (from E1)
#pragma unroll 1
  for (int s = 0; s < 16; ++s) {
    int k0 = 4 * s + kh;
#pragma unroll
    for (int t = 0; t < 5; ++t) {
      int r = rb + t * 16 + ml;
      v2f a = {E1[r][k0], E1[r][k0 + 1]};
      acc[t] = wmma_f32_16x16x4(a, wf[s], acc[t]);
    }
  }
// K in [64,128): edge = center (broadcast per point from Ce)
#pragma unroll 1
  for (int s = 16; s < 32; ++s) {
    int k0 = 4 * s + kh - 64;
#pragma unroll
    for (int t = 0; t < 5; ++t) {
      v2f a = {Ce[pl[t]][k0], Ce[pl[t]][k0 + 1]};
      acc[t] = wmma_f32_16x16x4(a, wf[s], acc[t]);
    }
  }

  // BN (inference) + LeakyReLU + max over K=20, all in registers
  const int o = o0 + ml;
  const float inv = gamma[o] * rsqrtf(rvar[o] + 1e-5f);
  const float shift = beta[o] - rmean[o] * inv;

#pragma unroll
  for (int q = 0; q < 4; ++q) {
    float m = -3.4e38f;
#pragma unroll
    for (int j = 0; j < KNN; ++j) {
      int lr = 20 * q + j;
      int t = lr >> 4, i = lr & 15;
      float v = acc[t][i & 7];
      float y = v * inv + shift;
      y = fmaxf(y, 0.2f * y);
      bool mine = ((i >> 3) == h);  // which lane-half owns this row
      m = mine ? fmaxf(m, y) : m;
    }
    float other = __shfl_xor(m, 16, 32);  // combine the two lane halves
    m = fmaxf(m, other);
    if (lane < 16) ymax[o][mhalf * 4 + q] = m;
  }
  __syncthreads();
#pragma unroll
  for (int e = 0; e < 2; ++e) {
    int flat = e * 256 + tid;
    int oo = flat >> 3, nl = flat & 7;
    out[((size_t)(b * OCH + oo)) * NTOT + n0 + nl] = ymax[oo][nl];
  }
}

// ---------------------------------------------------------------------------
extern "C" void kernel_launch(void* const* d_in, const int* in_sizes, int n_in,
                              void* d_out, int out_size, void* d_ws, size_t ws_size,
                              hipStream_t stream) {
  (void)in_sizes; (void)n_in; (void)out_size; (void)ws_size;
  const float* x = (const float*)d_in[0];
  const float* W = (const float*)d_in[1];
  const float* gamma = (const float*)d_in[2];
  const float* beta = (const float*)d_in[3];
  const float* rmean = (const float*)d_in[4];
  const float* rvar = (const float*)d_in[5];
  float* out = (float*)d_out;

  char* ws = (char*)d_ws;
  float* xt = (float*)ws;                                     // 8 MiB
  float* xx = (float*)(ws + (size_t)BATCH * NTOT * CIN * 4);  // 128 KiB
  int* idx = (int*)(ws + (size_t)BATCH * NTOT * CIN * 4 + (size_t)BATCH * NTOT * 4);  // 2.5 MiB

  k0_transpose_norm<<<dim3(NTOT / 256, BATCH), 256, 0, stream>>>(x, xt, xx);
  k1_knn<<<dim3(NTOT / 64, BATCH), 256, 0, stream>>>(xt, xx, idx);
  k2_edgeconv<<<dim3(NTOT / 8, BATCH), 256, 0, stream>>>(xt, idx, W, gamma, beta, rmean, rvar, out);
}